// MultitaskLossA_81999515615828
// MI455X (gfx1250) — compile-verified
//
#include <hip/hip_runtime.h>
#include <math.h>

typedef __attribute__((ext_vector_type(2))) float v2f;
typedef __attribute__((ext_vector_type(4))) float v4f;
typedef __attribute__((ext_vector_type(8))) float v8f;

#define MTL_BLOCK 256
#define MTL_GRID  1024

// Cross-lane (wave32) sum using V_WMMA_F32_16X16X4_F32.
// A-matrix 16x4 f32 layout (ISA 7.12.2): VGPR0 lanes0-15 = A[m,0], lanes16-31 = A[m,2];
// VGPR1 lanes0-15 = A[m,1], lanes16-31 = A[m,3]. With B = all-ones (layout independent),
// D[m,n] = A[m,0]+A[m,1]+A[m,2]+A[m,3] = p_m + p_{m+16}.
// D layout: VGPR j, lanes0-15 hold row M=j, lanes16-31 hold row M=j+8, so summing the 8
// D VGPRs gives rows 0..7 (lanes<16) / rows 8..15 (lanes>=16); one xor-16 shuffle totals it.
__device__ __forceinline__ float wave_reduce_wmma(float p) {
    v2f a; a.x = p; a.y = 0.0f;
    v2f bones; bones.x = 1.0f; bones.y = 1.0f;
    v8f c = {};
    v8f d = __builtin_amdgcn_wmma_f32_16x16x4_f32(
        /*neg_a=*/false, a, /*neg_b=*/false, bones,
        /*c_mod=*/(short)0, c, /*reuse_a=*/false, /*reuse_b=*/false);
    float s = d[0] + d[1] + d[2] + d[3] + d[4] + d[5] + d[6] + d[7];
    s += __shfl_xor(s, 16, 32);
    return s; // all lanes hold the full 32-lane sum
}

// Numerically stable BCE-with-logits using single-instruction trans ops:
// max(x,0) - x*y + log(1 + exp(-|x|));  exp arg <= 0 so no overflow, and the
// log argument is in (1,2] where v_log_f32 is well-conditioned (abs err ~1e-7).
__device__ __forceinline__ float bce_logits(float x, float y) {
    return fmaxf(x, 0.0f) - x * y + __logf(1.0f + __expf(-fabsf(x)));
}

__global__ void mtl_partial_kernel(const float* __restrict__ xb,   // [B]
                                   const float* __restrict__ xt,   // [B,5]
                                   const float* __restrict__ xs,   // [B,16]
                                   const float* __restrict__ yb,   // [B]
                                   const float* __restrict__ yt,   // [B,5]
                                   const int*  __restrict__ ysrc,  // [B]
                                   float* __restrict__ partials,   // [gridDim.x]
                                   int B) {
    const int tid    = blockIdx.x * blockDim.x + threadIdx.x;
    const int stride = gridDim.x * blockDim.x;

    float acc = 0.0f;
    for (int b = tid; b < B; b += stride) {
        // ---- task A: binary BCE ----
        const float x  = __builtin_nontemporal_load(xb + b);
        const float yv = __builtin_nontemporal_load(yb + b);
        float contrib = bce_logits(x, yv);

        // ---- task B: type BCE (mean over B*T -> weight 1/T here, 1/B at the end) ----
        const float* tp = xt + (size_t)b * 5;
        const float* tq = yt + (size_t)b * 5;
        float tv[5], tw[5];
#pragma unroll
        for (int t = 0; t < 5; ++t) {
            tv[t] = __builtin_nontemporal_load(tp + t);
            tw[t] = __builtin_nontemporal_load(tq + t);
        }
        float tsum = 0.0f;
#pragma unroll
        for (int t = 0; t < 5; ++t) tsum += bce_logits(tv[t], tw[t]);
        contrib += 0.2f * tsum;

        // ---- task C: source cross-entropy over 16 classes ----
        const v4f* sp = reinterpret_cast<const v4f*>(xs + (size_t)b * 16);
        v4f s0 = __builtin_nontemporal_load(sp + 0);
        v4f s1 = __builtin_nontemporal_load(sp + 1);
        v4f s2 = __builtin_nontemporal_load(sp + 2);
        v4f s3 = __builtin_nontemporal_load(sp + 3);
        float v[16] = { s0[0], s0[1], s0[2], s0[3],
                        s1[0], s1[1], s1[2], s1[3],
                        s2[0], s2[1], s2[2], s2[3],
                        s3[0], s3[1], s3[2], s3[3] };
        const int idx = __builtin_nontemporal_load(ysrc + b);
        float m = v[0];
        int   ai = 0;                       // first-occurrence argmax (jnp.argmax semantics)
        float xsel = (idx == 0) ? v[0] : 0.0f;
#pragma unroll
        for (int i = 1; i < 16; ++i) {
            if (v[i] > m) { m = v[i]; ai = i; }
            if (i == idx) xsel = v[i];
        }
        float es = 0.0f;
#pragma unroll
        for (int i = 0; i < 16; ++i) es += __expf(v[i] - m);   // args <= 0, underflow -> 0
        contrib += m + __logf(es) - xsel;   // lse - x[idx] = -log_softmax[idx]; es in [1,16]

        // ---- consistency XOR terms (ALPHA = 1) ----
        const bool bc = x     > 0.0f;
        const bool tc = tv[0] > 0.0f;
        const bool sc = ai > 0;
        contrib += (bc != tc ? 1.0f : 0.0f) + (bc != sc ? 1.0f : 0.0f);

        acc += contrib;
    }

    // wave-level reduce on the CDNA5 matrix pipe (EXEC is all-ones here)
    const float wsum = wave_reduce_wmma(acc);

    __shared__ float lds[MTL_BLOCK / 32];
    const int lane = threadIdx.x & 31;
    const int wid  = threadIdx.x >> 5;
    if (lane == 0) lds[wid] = wsum;
    __syncthreads();
    if (threadIdx.x == 0) {
        float s = 0.0f;
#pragma unroll
        for (int w = 0; w < MTL_BLOCK / 32; ++w) s += lds[w];
        partials[blockIdx.x] = s;
    }
}

__global__ void mtl_final_kernel(const float* __restrict__ partials, int n,
                                 float* __restrict__ out, float invB) {
    float acc = 0.0f;
    for (int i = threadIdx.x; i < n; i += 32) acc += partials[i];
    const float s = wave_reduce_wmma(acc);
    if (threadIdx.x == 0) out[0] = s * invB;
}

extern "C" void kernel_launch(void* const* d_in, const int* in_sizes, int n_in,
                              void* d_out, int out_size, void* d_ws, size_t ws_size,
                              hipStream_t stream) {
    const float* xb   = (const float*)d_in[0]; // y_pred_binary [B]
    const float* xt   = (const float*)d_in[1]; // y_pred_type   [B,5]
    const float* xs   = (const float*)d_in[2]; // y_pred_source [B,16]
    const float* yb   = (const float*)d_in[3]; // y_true_binary [B]
    const float* yt   = (const float*)d_in[4]; // y_true_type   [B,5]
    const int*   ysrc = (const int*)  d_in[5]; // y_true_source [B]

    const int B = in_sizes[0];
    float* partials = (float*)d_ws;            // MTL_GRID floats = 4 KB
    float* out      = (float*)d_out;

    mtl_partial_kernel<<<MTL_GRID, MTL_BLOCK, 0, stream>>>(
        xb, xt, xs, yb, yt, ysrc, partials, B);
    mtl_final_kernel<<<1, 32, 0, stream>>>(partials, MTL_GRID, out, 1.0f / (float)B);
}